// HypAgg_81887846466067
// MI455X (gfx1250) — compile-verified
//
#include <hip/hip_runtime.h>
#include <math.h>

#define NN 1024
#define DD 64
#define MIN_NORM 1e-15f
#define WAVES 4

typedef float v2f __attribute__((ext_vector_type(2)));
typedef float v8f __attribute__((ext_vector_type(8)));

// Accurate artanh (used in the O(N) kernels): matches reference clip + log1p form.
__device__ __forceinline__ float artanh_f(float z) {
    z = fminf(fmaxf(z, -1.0f + 1e-7f), 1.0f - 1e-7f);
    return 0.5f * (log1pf(z) - log1pf(-z));
}

// Fast artanh for the O(N^2) pair pass: z in [0, 1-1e-7]. Uses v_log_f32/v_rcp_f32
// (~1 ulp). Low-z relative error is harmless: near-coincident pairs have the same
// gamma on both alpha and beta terms, and their contributions cancel as x_j -> x_i.
__device__ __forceinline__ float artanh_fast(float z) {
    z = fminf(z, 1.0f - 1e-7f);
    return 0.5f * __logf((1.0f + z) * __builtin_amdgcn_rcpf(1.0f - z));
}

// ---------------------------------------------------------------------------
// Kernel 1: per-row stats. x2[i]=|x_i|^2, left[i]=s0*<x_i,w0>+b, right[i]=s0*<x_i,w1>
// ---------------------------------------------------------------------------
__global__ __launch_bounds__(256)
void hypagg_row_stats(const float* __restrict__ x, const float* __restrict__ r,
                      const float* __restrict__ w_att, const float* __restrict__ b_att,
                      float* __restrict__ x2ws, float* __restrict__ leftws,
                      float* __restrict__ rightws) {
    const int wave = threadIdx.x >> 5;
    const int lane = threadIdx.x & 31;
    const int row  = blockIdx.x * 8 + wave;
    const float* xr = x + (size_t)row * DD;
    float v0 = xr[lane], v1 = xr[lane + 32];
    float s2 = v0 * v0 + v1 * v1;
    float dl = v0 * w_att[lane] + v1 * w_att[lane + 32];
    float dr = v0 * w_att[DD + lane] + v1 * w_att[DD + lane + 32];
#pragma unroll
    for (int m = 1; m < 32; m <<= 1) {
        s2 += __shfl_xor(s2, m, 32);
        dl += __shfl_xor(dl, m, 32);
        dr += __shfl_xor(dr, m, 32);
    }
    if (lane == 0) {
        const float c  = 1.0f / (r[0] * r[0]);
        const float sc = sqrtf(c);
        const float nrm = sqrtf(fmaxf(s2, MIN_NORM));
        const float s0  = artanh_f(sc * nrm) / (sc * nrm);
        x2ws[row]    = s2;
        leftws[row]  = s0 * dl + b_att[0];
        rightws[row] = s0 * dr;
    }
}

// ---------------------------------------------------------------------------
// Kernel 2: fused pairwise kernel. 4 waves per block; each block owns a 16-row
// i-tile, waves split the 64 j-tiles (16 each). Per j-tile per wave:
//   G tile  = x_i(16xD) @ x_j^T(Dx16)  -> 16x v_wmma_f32_16x16x4_f32
//   pair scalars in accumulator layout (rcp/rsq based, no IEEE divides)
//   S tiles += Bt(16x16) @ x_j(16x64)  -> 16x v_wmma_f32_16x16x4_f32
// Final deterministic LDS reduction across the 4 waves.
// ---------------------------------------------------------------------------
__global__ __launch_bounds__(32 * WAVES)
void hypagg_pair(const float* __restrict__ x, const float* __restrict__ adj,
                 const float* __restrict__ r, const float* __restrict__ x2ws,
                 const float* __restrict__ leftws, const float* __restrict__ rightws,
                 float* __restrict__ supp) {
    __shared__ float bt_lds[WAVES][16 * 17];   // per-wave transpose staging (padded)
    __shared__ float red_acc[WAVES][16 * 64];  // per-wave partial S tiles (16 KB)
    __shared__ float red_asum[WAVES][16];
    __shared__ float asum_tot[16];

    const int tid  = threadIdx.x;
    const int wave = tid >> 5;
    const int lane = tid & 31;
    const int nidx = lane & 15;                // column within tile
    const int hi   = lane >> 4;                // half-wave select
    const int i0   = blockIdx.x * 16;

    const float c  = 1.0f / (r[0] * r[0]);
    const float sc = sqrtf(c);
    const float inv_sc = __builtin_amdgcn_rcpf(sc);

    // A operand (resident): rows i0..i0+15 of x as 16 K=4 chunks.
    // ISA A-layout 16x4 f32: lane m holds (K=2*hi, K=2*hi+1) -> one float2 each.
    v2f a[16];
    {
        const float* xa = x + (size_t)(i0 + nidx) * DD + 2 * hi;
#pragma unroll
        for (int kk = 0; kk < 16; ++kk) a[kk] = *(const v2f*)(xa + 4 * kk);
    }
    // Per-lane row scalars and hoisted j-loop invariants (m = v + 8*hi)
    float x2i[8], lefti[8], bnum[8], gfac[8], ccx2i[8];
#pragma unroll
    for (int v = 0; v < 8; ++v) {
        x2i[v]   = x2ws[i0 + v + 8 * hi];
        lefti[v] = leftws[i0 + v + 8 * hi];
        bnum[v]  = 1.0f - c * x2i[v];                    // beta numerator (unclipped)
        gfac[v]  = fmaxf(bnum[v], MIN_NORM) * inv_sc;    // 2/(sqrt_c*lam_i)
        ccx2i[v] = c * c * x2i[v];
    }

    v8f acc0 = {}, acc1 = {}, acc2 = {}, acc3 = {};
    float asum[8];
#pragma unroll
    for (int v = 0; v < 8; ++v) asum[v] = 0.0f;

    for (int jt = wave; jt < NN / 16; jt += WAVES) {
        const int j0 = jt * 16;
        // ---- Gram tile via f32 WMMA (exact f32 dot products) ----
        const float* xb = x + (size_t)(j0 + nidx) * DD + 2 * hi;
        v8f g = {};
#pragma unroll
        for (int kk = 0; kk < 16; ++kk) {
            v2f b = *(const v2f*)(xb + 4 * kk);
            g = __builtin_amdgcn_wmma_f32_16x16x4_f32(false, a[kk], false, b,
                                                      (short)0, g, false, false);
        }

        const float y2  = x2ws[j0 + nidx];
        const float rj  = rightws[j0 + nidx];
        const float cy2 = c * y2;

        float bt[8];
#pragma unroll
        for (int v = 0; v < 8; ++v) {
            const int i = i0 + v + 8 * hi;
            const float xy  = g[v];
            const float adv = adj[(size_t)i * NN + j0 + nidx];
            const float t   = fmaf(-2.0f * c, xy, 1.0f);
            const float den = fmaxf(fmaf(ccx2i[v], y2, t), MIN_NORM);
            const float inv_den = __builtin_amdgcn_rcpf(den);      // ~1 ulp
            const float alpha = -(t + cy2) * inv_den;
            const float beta  = bnum[v] * inv_den;
            const float sub2  = alpha * (alpha * x2i[v] + 2.0f * beta * xy)
                              + beta * beta * y2;
            const float s     = fmaxf(sub2, MIN_NORM);
            const float rsn   = __builtin_amdgcn_rsqf(s);          // 1/|sub|, ~1 ulp
            const float subn  = s * rsn;
            const float gamma = gfac[v] * artanh_fast(sc * subn) * rsn;
            const float sig   = __builtin_amdgcn_rcpf(1.0f + __expf(-(lefti[v] + rj)));
            const float w     = adv * sig * gamma;
            asum[v] += w * alpha;
            bt[v]    = w * beta;
        }

        // ---- D-layout -> A-layout transpose of the Bt tile through LDS ----
        __syncthreads();                       // uniform trip counts across waves
#pragma unroll
        for (int v = 0; v < 8; ++v)
            bt_lds[wave][(v + 8 * hi) * 17 + nidx] = bt[v];
        __syncthreads();
        v2f ab[4];
#pragma unroll
        for (int kk = 0; kk < 4; ++kk) {
            ab[kk].x = bt_lds[wave][nidx * 17 + 4 * kk + 2 * hi];
            ab[kk].y = bt_lds[wave][nidx * 17 + 4 * kk + 2 * hi + 1];
        }
        __syncthreads();

        // ---- S[16x64] += Bt(16x16) @ x_j(16x64), 4 K-chunks x 4 col-tiles ----
#pragma unroll
        for (int kk = 0; kk < 4; ++kk) {
            const float* xr0 = x + (size_t)(j0 + 4 * kk + 2 * hi) * DD + nidx;
            const float* xr1 = xr0 + DD;
            v2f b0, b1, b2, b3;
            b0.x = xr0[0];  b0.y = xr1[0];
            b1.x = xr0[16]; b1.y = xr1[16];
            b2.x = xr0[32]; b2.y = xr1[32];
            b3.x = xr0[48]; b3.y = xr1[48];
            acc0 = __builtin_amdgcn_wmma_f32_16x16x4_f32(false, ab[kk], false, b0, (short)0, acc0, false, false);
            acc1 = __builtin_amdgcn_wmma_f32_16x16x4_f32(false, ab[kk], false, b1, (short)0, acc1, false, false);
            acc2 = __builtin_amdgcn_wmma_f32_16x16x4_f32(false, ab[kk], false, b2, (short)0, acc2, false, false);
            acc3 = __builtin_amdgcn_wmma_f32_16x16x4_f32(false, ab[kk], false, b3, (short)0, acc3, false, false);
        }
    }

    // In-wave reduction of asum over the 16 columns of each half-wave
#pragma unroll
    for (int v = 0; v < 8; ++v) {
        float s = asum[v];
        s += __shfl_xor(s, 1, 32);
        s += __shfl_xor(s, 2, 32);
        s += __shfl_xor(s, 4, 32);
        s += __shfl_xor(s, 8, 32);
        asum[v] = s;
    }

    // Stage per-wave partials to LDS (fixed order -> deterministic)
#pragma unroll
    for (int v = 0; v < 8; ++v) {
        const int row = v + 8 * hi;
        if (nidx == 0) red_asum[wave][row] = asum[v];
        red_acc[wave][row * 64 + nidx]      = acc0[v];
        red_acc[wave][row * 64 + nidx + 16] = acc1[v];
        red_acc[wave][row * 64 + nidx + 32] = acc2[v];
        red_acc[wave][row * 64 + nidx + 48] = acc3[v];
    }
    __syncthreads();
    if (tid < 16) {
        float s = 0.0f;
#pragma unroll
        for (int w = 0; w < WAVES; ++w) s += red_asum[w][tid];
        asum_tot[tid] = s;
    }
    __syncthreads();

    // Combine: support_t[i] = (sum_w S_w)[i] + asum_i * x_i ; 8 elements/thread
#pragma unroll
    for (int e = 0; e < 8; ++e) {
        const int flat = tid + 128 * e;            // 0..1023, coalesced
        const int row = flat >> 6, col = flat & 63;
        float s = red_acc[0][flat] + red_acc[1][flat]
                + red_acc[2][flat] + red_acc[3][flat];
        const int i = i0 + row;
        supp[(size_t)i * DD + col] = s + asum_tot[row] * x[(size_t)i * DD + col];
    }
}

// ---------------------------------------------------------------------------
// Kernel 3: out = proj(expmap(x, support_t, c), c). One wave per row.
// ---------------------------------------------------------------------------
__global__ __launch_bounds__(256)
void hypagg_finish(const float* __restrict__ x, const float* __restrict__ r,
                   const float* __restrict__ x2ws, const float* __restrict__ supp,
                   float* __restrict__ out) {
    const int wave = threadIdx.x >> 5;
    const int lane = threadIdx.x & 31;
    const int row  = blockIdx.x * 8 + wave;
    const float* xr = x + (size_t)row * DD;
    const float* ur = supp + (size_t)row * DD;
    float x0 = xr[lane], x1 = xr[lane + 32];
    float u0 = ur[lane], u1 = ur[lane + 32];
    float u2 = u0 * u0 + u1 * u1;
    float xu = x0 * u0 + x1 * u1;
#pragma unroll
    for (int m = 1; m < 32; m <<= 1) {
        u2 += __shfl_xor(u2, m, 32);
        xu += __shfl_xor(xu, m, 32);
    }
    const float c  = 1.0f / (r[0] * r[0]);
    const float sc = sqrtf(c);
    const float x2 = x2ws[row];
    const float un  = sqrtf(fmaxf(u2, MIN_NORM));
    const float lam = 2.0f / fmaxf(1.0f - c * x2, MIN_NORM);
    const float t   = tanhf(0.5f * sc * lam * un) / (sc * un);   // second = t*u
    const float s2s = t * t * u2;
    const float xs  = t * xu;
    const float num1 = 1.0f + 2.0f * c * xs + c * s2s;
    const float num2 = 1.0f - c * x2;
    const float den  = fmaxf(1.0f + 2.0f * c * xs + c * c * x2 * s2s, MIN_NORM);
    float y0 = (num1 * x0 + num2 * (t * u0)) / den;
    float y1 = (num1 * x1 + num2 * (t * u1)) / den;
    float y2s = y0 * y0 + y1 * y1;
#pragma unroll
    for (int m = 1; m < 32; m <<= 1) y2s += __shfl_xor(y2s, m, 32);
    const float yn   = sqrtf(fmaxf(y2s, MIN_NORM));
    const float maxn = (1.0f - 4e-3f) / sc;
    const float scl  = (yn > maxn) ? (maxn / yn) : 1.0f;
    out[(size_t)row * DD + lane]      = y0 * scl;
    out[(size_t)row * DD + lane + 32] = y1 * scl;
}

extern "C" void kernel_launch(void* const* d_in, const int* in_sizes, int n_in,
                              void* d_out, int out_size, void* d_ws, size_t ws_size,
                              hipStream_t stream) {
    const float* x     = (const float*)d_in[0];   // [1024,64]
    const float* adj   = (const float*)d_in[1];   // [1024,1024]
    const float* r     = (const float*)d_in[2];   // [1]
    const float* w_att = (const float*)d_in[3];   // [128,1]
    const float* b_att = (const float*)d_in[4];   // [1]
    float* out = (float*)d_out;

    float* ws      = (float*)d_ws;   // needs (3*1024 + 1024*64)*4 = 268 KB
    float* x2ws    = ws;
    float* leftws  = ws + NN;
    float* rightws = ws + 2 * NN;
    float* supp    = ws + 3 * NN;

    hypagg_row_stats<<<NN / 8, 256, 0, stream>>>(x, r, w_att, b_att, x2ws, leftws, rightws);
    hypagg_pair<<<NN / 16, 32 * WAVES, 0, stream>>>(x, adj, r, x2ws, leftws, rightws, supp);
    hypagg_finish<<<NN / 8, 256, 0, stream>>>(x, r, x2ws, supp, out);
}